// PtModule_76166950027835
// MI455X (gfx1250) — compile-verified
//
#include <hip/hip_runtime.h>

// Elementwise: out = sqrt((x + 5) * 2 - 3), 8192x8192 fp32.
// Bandwidth-bound: 512 MB traffic -> ~22 us floor at 23.3 TB/s.
// This revision minimizes non-VMEM overhead: 32-bit block-tile indexing so
// the 4 NT b128 loads/stores share one VGPR address + immediate offsets,
// and an unguarded fast path for full tiles (no exec-mask churn).

typedef float v4f __attribute__((ext_vector_type(4)));

#define THREADS 256                    // 8 wave32s per block
#define UNROLL  4                      // 4 b128 loads in flight per thread
#define TILE    (THREADS * UNROLL)     // float4s per block = 1024

__device__ __forceinline__ float fused_op(float x) {
    // Mirror the reference's rounding: RN(x+5), *2 (exact), -3 (exact), sqrt.
    return __builtin_amdgcn_sqrtf((x + 5.0f) * 2.0f - 3.0f);
}

__global__ __launch_bounds__(THREADS)
void sqrt_stream_kernel(const float* __restrict__ in,
                        float* __restrict__ out,
                        unsigned int nvec,      // number of float4 chunks
                        long long n) {          // total elements
    const v4f* __restrict__ in4  = (const v4f*)in;
    v4f*       __restrict__ out4 = (v4f*)out;

    const unsigned int block_base = blockIdx.x * TILE;
    const unsigned int base       = block_base + threadIdx.x;

    if (block_base + TILE <= nvec) {
        // ---- Fast path: full tile, no guards. ----
        // 4 loads share one address register; offsets u*THREADS*16B = u*4096
        // become instruction immediates -> single s_clause of b128 NT loads.
        v4f v[UNROLL];
#pragma unroll
        for (int u = 0; u < UNROLL; ++u)
            v[u] = __builtin_nontemporal_load(&in4[base + u * THREADS]);

#pragma unroll
        for (int u = 0; u < UNROLL; ++u) {
            v4f x = v[u], r;
#pragma unroll
            for (int k = 0; k < 4; ++k) r[k] = fused_op(x[k]);
            __builtin_nontemporal_store(r, &out4[base + u * THREADS]);
        }
    } else {
        // ---- Slow path: at most one partial block per launch. ----
#pragma unroll
        for (int u = 0; u < UNROLL; ++u) {
            const unsigned int i = base + u * THREADS;
            if (i < nvec) {
                v4f x = __builtin_nontemporal_load(&in4[i]);
                v4f r;
#pragma unroll
                for (int k = 0; k < 4; ++k) r[k] = fused_op(x[k]);
                __builtin_nontemporal_store(r, &out4[i]);
            }
        }
    }

    // ---- Scalar tail for n % 4 (first threads of block 0 only). ----
    const unsigned int gtid = blockIdx.x * THREADS + threadIdx.x;
    const long long tail_base = (long long)nvec << 2;
    const long long t = tail_base + gtid;
    if (blockIdx.x == 0 && t < n) {
        out[t] = fused_op(in[t]);
    }
}

extern "C" void kernel_launch(void* const* d_in, const int* in_sizes, int n_in,
                              void* d_out, int out_size, void* d_ws, size_t ws_size,
                              hipStream_t stream) {
    const float* in  = (const float*)d_in[0];
    float*       out = (float*)d_out;
    const long long n = (long long)in_sizes[0];   // 67,108,864 for 8192x8192

    const unsigned int nvec = (unsigned int)(n >> 2);   // 16,777,216 float4s
    long long blocks = ((long long)nvec + TILE - 1) / TILE;
    if (blocks < 1) blocks = 1;
    // 8192x8192: 16384 blocks -> 131072 wave32s, saturates every WGP.

    sqrt_stream_kernel<<<(unsigned)blocks, THREADS, 0, stream>>>(in, out, nvec, n);
}